// posnegECEbinsLoss_47923245089178
// MI455X (gfx1250) — compile-verified
//
#include <hip/hip_runtime.h>
#include <hip/hip_bf16.h>
#include <stdint.h>

#define NBINS 15
#define NCLS  64
#define TILE_ROWS 64
#define THREADS 256
#define WAVES_PER_BLOCK (THREADS / 32)
#define ROWS_PER_WAVE   (TILE_ROWS / WAVES_PER_BLOCK)
#define SEGS (NCLS * NBINS)                 /* 960 */
#define TILE_BYTES (TILE_ROWS * NCLS * 4)   /* 16 KB */
#define CHUNKS_PER_THREAD (TILE_BYTES / (THREADS * 16))  /* 4 */

typedef int v4i __attribute__((vector_size(16)));
typedef __attribute__((address_space(1))) v4i g1_v4i;   // global (AS1) int4
typedef __attribute__((address_space(3))) v4i lds_v4i;  // LDS   (AS3) int4

// ---------------- CDNA5 async global->LDS copy (ASYNCcnt path) ----------------
__device__ __forceinline__ void async_copy_b128(const void* gptr, void* lptr) {
#if __has_builtin(__builtin_amdgcn_global_load_async_to_lds_b128)
  __builtin_amdgcn_global_load_async_to_lds_b128((g1_v4i*)gptr, (lds_v4i*)lptr, 0, 0);
#else
  uint32_t loff = (uint32_t)(uintptr_t)(__attribute__((address_space(3))) char*)lptr;
  asm volatile("global_load_async_to_lds_b128 %0, %1, off"
               :: "v"(loff), "v"((uint64_t)(uintptr_t)gptr)
               : "memory");
#endif
}

__device__ __forceinline__ void wait_async_le4() {
#if __has_builtin(__builtin_amdgcn_s_wait_asynccnt)
  __builtin_amdgcn_s_wait_asynccnt(4);
#else
  asm volatile("s_wait_asynccnt 4" ::: "memory");
#endif
}
__device__ __forceinline__ void wait_async_0() {
#if __has_builtin(__builtin_amdgcn_s_wait_asynccnt)
  __builtin_amdgcn_s_wait_asynccnt(0);
#else
  asm volatile("s_wait_asynccnt 0" ::: "memory");
#endif
}

__device__ __forceinline__ void stage_tile(const float* __restrict__ logits, int tile,
                                           float* lbuf, int tid) {
  const char* g = (const char*)logits + (size_t)tile * TILE_BYTES;
  char* l = (char*)lbuf;
#pragma unroll
  for (int k = 0; k < CHUNKS_PER_THREAD; ++k) {
    int off = (tid + k * THREADS) * 16;   // linear 16B chunks: coalesced 512B per wave op
    async_copy_b128(g + off, l + off);
  }
}

// ---------------- binning (matches searchsorted(uppers, p, 'left') clipped) ----------------
__device__ __forceinline__ void bins_accum(float p, int c, int lab,
                                           float* s_cnt, float* s_conf, float* s_acc) {
  if (!(p > 0.0f)) return;                 // 'valid' mask in reference
  const float inv15 = 1.0f / 15.0f;        // linspace step in f32
  int b = (int)ceilf(p * 15.0f) - 1;
  b = b < 0 ? 0 : (b > NBINS - 1 ? NBINS - 1 : b);
  while (b > 0 && (float)b * inv15 >= p) --b;              // uppers[b-1] >= p
  while (b < NBINS - 1 && (float)(b + 1) * inv15 < p) ++b; // uppers[b]   <  p
  int idx = c * NBINS + b;
  atomicAdd(&s_cnt[idx], 1.0f);            // ds_add_f32
  atomicAdd(&s_conf[idx], p);
  if (lab == c) atomicAdd(&s_acc[idx], 1.0f);
}

// One wave32 owns one row: lane holds columns 2*lane, 2*lane+1.
__device__ __forceinline__ void process_row(float x0, float x1, int lane, int lab,
                                            float* s_cnt, float* s_conf, float* s_acc) {
  float m = fmaxf(x0, x1);
#pragma unroll
  for (int off = 16; off >= 1; off >>= 1) m = fmaxf(m, __shfl_xor(m, off, 32));
  float e0 = __expf(x0 - m);
  float e1 = __expf(x1 - m);
  float s = e0 + e1;
#pragma unroll
  for (int off = 16; off >= 1; off >>= 1) s += __shfl_xor(s, off, 32);
  float inv = 1.0f / s;
  bins_accum(e0 * inv, 2 * lane, lab, s_cnt, s_conf, s_acc);
  bins_accum(e1 * inv, 2 * lane + 1, lab, s_cnt, s_conf, s_acc);
}

// ---------------- kernels ----------------
__global__ void zero_ws_kernel(float* __restrict__ ws) {
  int i = blockIdx.x * blockDim.x + threadIdx.x;
  if (i < 3 * SEGS) ws[i] = 0.0f;
}

__global__ void __launch_bounds__(THREADS)
ece_accum_kernel(const float* __restrict__ logits, const int* __restrict__ labels,
                 float* __restrict__ ws, int nRows, int numTiles) {
  __shared__ float s_tile[2][TILE_ROWS * NCLS];   // 32 KB double buffer
  __shared__ float s_cnt[SEGS];
  __shared__ float s_conf[SEGS];
  __shared__ float s_acc[SEGS];

  const int tid  = threadIdx.x;
  const int wave = tid >> 5;
  const int lane = tid & 31;

  for (int i = tid; i < SEGS; i += THREADS) { s_cnt[i] = 0.f; s_conf[i] = 0.f; s_acc[i] = 0.f; }

  // prologue: stage first tile for this block
  if ((int)blockIdx.x < numTiles) stage_tile(logits, blockIdx.x, s_tile[0], tid);

  int buf = 0;
  for (int t = blockIdx.x; t < numTiles; t += gridDim.x, buf ^= 1) {
    const int nt = t + (int)gridDim.x;
    const bool hasNext = nt < numTiles;
    if (hasNext) stage_tile(logits, nt, s_tile[buf ^ 1], tid);  // overlap DMA with compute
    if (hasNext) wait_async_le4(); else wait_async_0();         // tile t's 4 asyncs retired
    __syncthreads();

    const float* tb = s_tile[buf];
    const int rowBase = t * TILE_ROWS;
#pragma unroll
    for (int r = 0; r < ROWS_PER_WAVE; ++r) {
      const int lr  = wave * ROWS_PER_WAVE + r;
      const int row = rowBase + lr;
      const float2 xv = *(const float2*)&tb[lr * NCLS + lane * 2];  // ds_load_b64
      const int lab = labels[row];
      process_row(xv.x, xv.y, lane, lab, s_cnt, s_conf, s_acc);
    }
    __syncthreads();   // all reads of s_tile[buf] done before it is re-staged
  }

  // remainder rows (N % 64 != 0 path; dead for this instance but kept general)
  const int remStart = numTiles * TILE_ROWS;
  if (remStart < nRows) {
    const int gw = blockIdx.x * WAVES_PER_BLOCK + wave;
    const int tw = gridDim.x * WAVES_PER_BLOCK;
    for (int row = remStart + gw; row < nRows; row += tw) {
      const float x0 = logits[(size_t)row * NCLS + lane * 2];
      const float x1 = logits[(size_t)row * NCLS + lane * 2 + 1];
      process_row(x0, x1, lane, labels[row], s_cnt, s_conf, s_acc);
    }
  }

  __syncthreads();
  for (int i = tid; i < SEGS; i += THREADS) {
    const float c = s_cnt[i];
    if (c != 0.0f) {
      atomicAdd(&ws[i], c);
      atomicAdd(&ws[SEGS + i], s_conf[i]);
      atomicAdd(&ws[2 * SEGS + i], s_acc[i]);
    }
  }
}

__global__ void finalize_kernel(const float* __restrict__ ws, float* __restrict__ out,
                                int nRows) {
  const int b = threadIdx.x;
  if (b >= NBINS) return;
  float ov = 0.0f, un = 0.0f;
  const float invN = 1.0f / (float)nRows;
  for (int c = 0; c < NCLS; ++c) {          // fixed order: deterministic
    const int i = c * NBINS + b;
    const float cnt = ws[i];
    if (cnt > 0.0f) {
      const float denom = fmaxf(cnt, 1.0f);
      const float gap = ws[SEGS + i] / denom - ws[2 * SEGS + i] / denom;
      const float contrib = fabsf(gap) * (cnt * invN);
      if (gap > 0.0f) ov += contrib; else un += contrib;   // gap<=0 -> under (matches ref)
    }
  }
  out[b]             = ov;
  out[NBINS + b]     = un;
  out[2 * NBINS + b] = (float)b * (1.0f / 15.0f);          // lowers = linspace(0,1,16)[:-1]
}

extern "C" void kernel_launch(void* const* d_in, const int* in_sizes, int n_in,
                              void* d_out, int out_size, void* d_ws, size_t ws_size,
                              hipStream_t stream) {
  const float* logits = (const float*)d_in[0];
  const int*   labels = (const int*)d_in[1];
  float* out = (float*)d_out;
  float* ws  = (float*)d_ws;

  const int nRows = in_sizes[1];            // N (labels count)
  const int numTiles = nRows / TILE_ROWS;

  zero_ws_kernel<<<(3 * SEGS + THREADS - 1) / THREADS, THREADS, 0, stream>>>(ws);

  int grid = numTiles > 0 ? (numTiles < 2048 ? numTiles : 2048) : 1;
  ece_accum_kernel<<<grid, THREADS, 0, stream>>>(logits, labels, ws, nRows, numTiles);

  finalize_kernel<<<1, 32, 0, stream>>>(ws, out, nRows);
}